// RecurrentDecoder_54571854463854
// MI455X (gfx1250) — compile-verified
//
#include <hip/hip_runtime.h>
#include <hip/hip_bf16.h>

// ---------------------------------------------------------------------------
// RecurrentDecoder for MI455X (gfx1250, wave32).
//   H=512, T=400 steps, S=400 frames, B=32.
// fc(h+enc) = fc(enc) + fc(h)  (fc linear, no bias) -> enc_proj = fc(enc)
// hoisted out of the scan (one 3.36 GMAC WMMA GEMM). Per-step work is then
// L2-bandwidth bound (enc_proj 26MB + enc 26MB swept once per step, both
// L2-resident on 192MB L2) with an online flash-softmax. GRU/mel GEMMs use
// V_WMMA_F32_16X16X4_F32 with 4 independent accumulators to break the
// WMMA->WMMA RAW chain (per-step GEMMs are latency-bound at M=32).
// One persistent kernel + device barrier avoids ~2000 launch overheads.
// Workspace: ~29 MB in d_ws.
// ---------------------------------------------------------------------------

#define HID     512
#define TSTEPS  400
#define SFRM    400
#define BATCH   32
#define NWG     128
#define TPB     256
#define WPB     8                 // waves per workgroup (wave32)
#define NWAVES  (NWG * WPB)       // 1024
#define NCHUNK  32                // s-chunks for flash softmax

typedef __attribute__((ext_vector_type(2))) float v2f;
typedef __attribute__((ext_vector_type(8))) float v8f;

#if __has_builtin(__builtin_amdgcn_tanhf)
#define TANHF(x) __builtin_amdgcn_tanhf(x)
#elif __has_builtin(__builtin_amdgcn_tanh_f32)
#define TANHF(x) __builtin_amdgcn_tanh_f32(x)
#else
#define TANHF(x) tanhf(x)
#endif

// ---- workspace layout (in floats) ----
#define OFF_ENCPROJ 0ul                                   // S*B*H
#define OFF_H       (OFF_ENCPROJ + (size_t)SFRM*BATCH*HID)// B*H
#define OFF_FCH     (OFF_H    + (size_t)BATCH*HID)        // B*H
#define OFF_CTX     (OFF_FCH  + (size_t)BATCH*HID)        // B*H
#define OFF_GI      (OFF_CTX  + (size_t)BATCH*HID)        // B*3H
#define OFF_GH      (OFF_GI   + (size_t)BATCH*3*HID)      // B*3H
#define OFF_PM      (OFF_GH   + (size_t)BATCH*3*HID)      // NCHUNK*B
#define OFF_PL      (OFF_PM   + (size_t)NCHUNK*BATCH)     // NCHUNK*B
#define OFF_PCTX    (OFF_PL   + (size_t)NCHUNK*BATCH)     // NCHUNK*B*H
#define OFF_BAR     (OFF_PCTX + (size_t)NCHUNK*BATCH*HID) // 2 uints

__device__ inline float sigf(float x) { return 1.0f / (1.0f + __expf(-x)); }

__device__ inline float wred_sum(float v) {
#pragma unroll
  for (int o = 16; o > 0; o >>= 1) v += __shfl_xor(v, o, 32);
  return v;
}
__device__ inline float wred_max(float v) {
#pragma unroll
  for (int o = 16; o > 0; o >>= 1) v = fmaxf(v, __shfl_xor(v, o, 32));
  return v;
}

// Device-wide sense-reversing barrier (persistent kernel; NWG co-resident WGs).
__device__ inline void grid_sync(unsigned* bar) {
  __syncthreads();
  if (threadIdx.x == 0) {
    __threadfence();
    unsigned g = __hip_atomic_load(&bar[1], __ATOMIC_RELAXED, __HIP_MEMORY_SCOPE_AGENT);
    unsigned a = __hip_atomic_fetch_add(&bar[0], 1u, __ATOMIC_ACQ_REL, __HIP_MEMORY_SCOPE_AGENT) + 1u;
    if (a == (unsigned)NWG) {
      __hip_atomic_store(&bar[0], 0u, __ATOMIC_RELAXED, __HIP_MEMORY_SCOPE_AGENT);
      __hip_atomic_fetch_add(&bar[1], 1u, __ATOMIC_RELEASE, __HIP_MEMORY_SCOPE_AGENT);
    } else {
      while (__hip_atomic_load(&bar[1], __ATOMIC_ACQUIRE, __HIP_MEMORY_SCOPE_AGENT) == g)
        __builtin_amdgcn_s_sleep(1);
    }
    __threadfence();
  }
  __syncthreads();
}

#define WMMA_F32X4(A, Bv, Cv) \
  __builtin_amdgcn_wmma_f32_16x16x4_f32(false, (A), false, (Bv), (short)0, (Cv), false, false)

// One 16x16 tile of C = X(MxK, ldx) * W(NxK, ldw)^T via V_WMMA_F32_16X16X4_F32.
// 4 independent accumulators (K step 16) break the WMMA->WMMA RAW chain.
// A 16x4: lanes 0-15 M rows {K=k0,k0+1}, lanes 16-31 {K=k0+2,k0+3}.
// B 4x16: same half split on K.  D: vgpr i -> row half*8+i, col = lane&15.
template <bool HAS_BIAS>
__device__ inline void gemm_tile_xwt(const float* __restrict__ X, int ldx,
                                     const float* __restrict__ W, int ldw,
                                     int K, int m0, int n0,
                                     float* __restrict__ C, long ldc,
                                     const float* __restrict__ bias) {
  const int lane = threadIdx.x & 31;
  const int half = lane >> 4;
  const int lm   = lane & 15;
  const float* xr = X + (size_t)(m0 + lm) * ldx + half * 2;
  const float* wr = W + (size_t)(n0 + lm) * ldw + half * 2;
  v8f z = {0.f, 0.f, 0.f, 0.f, 0.f, 0.f, 0.f, 0.f};
  v8f acc0 = z, acc1 = z, acc2 = z, acc3 = z;
  for (int k0 = 0; k0 < K; k0 += 16) {
    acc0 = WMMA_F32X4(*(const v2f*)(xr + k0 + 0),  *(const v2f*)(wr + k0 + 0),  acc0);
    acc1 = WMMA_F32X4(*(const v2f*)(xr + k0 + 4),  *(const v2f*)(wr + k0 + 4),  acc1);
    acc2 = WMMA_F32X4(*(const v2f*)(xr + k0 + 8),  *(const v2f*)(wr + k0 + 8),  acc2);
    acc3 = WMMA_F32X4(*(const v2f*)(xr + k0 + 12), *(const v2f*)(wr + k0 + 12), acc3);
  }
  v8f acc = (acc0 + acc1) + (acc2 + acc3);
  if (HAS_BIAS) {
    const float bv = bias[n0 + lm];
#pragma unroll
    for (int i = 0; i < 8; ++i)
      C[(size_t)(m0 + half * 8 + i) * ldc + (n0 + lm)] = acc[i] + bv;
  } else {
#pragma unroll
    for (int i = 0; i < 8; ++i)
      C[(size_t)(m0 + half * 8 + i) * ldc + (n0 + lm)] = acc[i];
  }
}

// A rows = concat [x | ctx] (K = 2H); x==null at t==0 (zero frame) -> the
// first H columns contribute nothing, so start K at H.
__device__ inline void gemm_tile_concat(const float* __restrict__ x,
                                        const float* __restrict__ ctx,
                                        const float* __restrict__ W,
                                        int m0, int n0,
                                        float* __restrict__ C, long ldc) {
  const int lane = threadIdx.x & 31;
  const int half = lane >> 4;
  const int lm   = lane & 15;
  const float* xr = x ? x + (size_t)(m0 + lm) * HID + half * 2 : nullptr;
  const float* cr = ctx + (size_t)(m0 + lm) * HID + half * 2;
  const float* wr = W + (size_t)(n0 + lm) * (2 * HID) + half * 2;
  v8f z = {0.f, 0.f, 0.f, 0.f, 0.f, 0.f, 0.f, 0.f};
  v8f acc0 = z, acc1 = z, acc2 = z, acc3 = z;
  const int kstart = x ? 0 : HID;                 // HID % 16 == 0
  for (int k0 = kstart; k0 < 2 * HID; k0 += 16) { // boundary never straddles
    const float* ar = (k0 < HID) ? xr + k0 : cr + (k0 - HID);
    acc0 = WMMA_F32X4(*(const v2f*)(ar + 0),  *(const v2f*)(wr + k0 + 0),  acc0);
    acc1 = WMMA_F32X4(*(const v2f*)(ar + 4),  *(const v2f*)(wr + k0 + 4),  acc1);
    acc2 = WMMA_F32X4(*(const v2f*)(ar + 8),  *(const v2f*)(wr + k0 + 8),  acc2);
    acc3 = WMMA_F32X4(*(const v2f*)(ar + 12), *(const v2f*)(wr + k0 + 12), acc3);
  }
  v8f acc = (acc0 + acc1) + (acc2 + acc3);
#pragma unroll
  for (int i = 0; i < 8; ++i)
    C[(size_t)(m0 + half * 8 + i) * ldc + (n0 + lm)] = acc[i];
}

__global__ void init_bar_kernel(unsigned* bar) { bar[0] = 0u; bar[1] = 0u; }

__global__ __launch_bounds__(TPB)
void recurrent_decoder_kernel(const float* __restrict__ dec_in,   // (T,B,H)
                              const float* __restrict__ enc,      // (S,B,H)
                              const float* __restrict__ fc_W,     // (H,H)
                              const float* __restrict__ att_w,    // (H)
                              const float* __restrict__ Wih,      // (3H,2H)
                              const float* __restrict__ Whh,      // (3H,H)
                              const float* __restrict__ spec_W,   // (H,H)
                              const float* __restrict__ spec_b,   // (H)
                              const float* __restrict__ gate_W,   // (H)
                              const float* __restrict__ gate_b,   // (1)
                              float* __restrict__ out,            // B*T*H mel + B*T gate
                              float* __restrict__ ws) {
  const int tid  = threadIdx.x;
  const int lane = tid & 31;
  const int wv   = tid >> 5;
  const int gw   = blockIdx.x * WPB + wv;       // global wave id [0, NWAVES)

  float*    encp = ws + OFF_ENCPROJ;
  float*    hbuf = ws + OFF_H;
  float*    fch  = ws + OFF_FCH;
  float*    ctx  = ws + OFF_CTX;
  float*    gi   = ws + OFF_GI;
  float*    gh   = ws + OFF_GH;
  float*    pm   = ws + OFF_PM;
  float*    pl   = ws + OFF_PL;
  float*    pctx = ws + OFF_PCTX;
  unsigned* bar  = (unsigned*)(ws + OFF_BAR);

  float* mel_base  = out;                                 // (B,T,H)
  float* gate_base = out + (size_t)BATCH * TSTEPS * HID;  // (B,T)

  // ---- pre-phase: h = 0; enc_proj = enc @ fc_W^T (hoisted out of scan) ----
  for (int i = blockIdx.x * TPB + tid; i < BATCH * HID; i += NWG * TPB)
    hbuf[i] = 0.f;
  {
    const int ntiles = (SFRM * BATCH / 16) * (HID / 16);  // 800*32 = 25600
    for (int tile = gw; tile < ntiles; tile += NWAVES) {
      const int mt = tile >> 5;        // 800 row tiles over (s*B+b)
      const int nt = tile & 31;        // 32 col tiles
      gemm_tile_xwt<false>(enc, HID, fc_W, HID, HID, mt * 16, nt * 16,
                           encp, HID, nullptr);
    }
  }
  grid_sync(bar);

  for (int t = 0; t <= TSTEPS; ++t) {
    // ---- Phase A: fc_h = h @ fc_W^T;  mel/gate outputs for step t-1 ----
    if (t < TSTEPS && gw < 64) {
      const int mt = gw & 1, nt = gw >> 1;
      gemm_tile_xwt<false>(hbuf, HID, fc_W, HID, HID, mt * 16, nt * 16,
                           fch, HID, nullptr);
    }
    if (t >= 1) {
      if (gw >= 64 && gw < 128) {      // mel = h2 @ spec_W^T + b -> out[b][t-1][:]
        const int g2 = gw - 64;
        const int mt = g2 & 1, nt = g2 >> 1;
        gemm_tile_xwt<true>(hbuf, HID, spec_W, HID, HID, mt * 16, nt * 16,
                            mel_base + (size_t)(t - 1) * HID, (long)TSTEPS * HID,
                            spec_b);
      }
      if (blockIdx.x == 16) {          // gate = h2 . gate_W + gate_b
#pragma unroll
        for (int q = 0; q < 4; ++q) {
          const int b = wv * 4 + q;
          float p = 0.f;
          for (int j = lane; j < HID; j += 32) p += hbuf[b * HID + j] * gate_W[j];
          p = wred_sum(p);
          if (lane == 0) gate_base[(size_t)b * TSTEPS + (t - 1)] = p + gate_b[0];
        }
      }
    }
    grid_sync(bar);
    if (t == TSTEPS) break;

    // ---- Phase B: flash-softmax sweep over s-chunks, all 1024 waves ----
    {
      const int b  = gw & 31;          // batch element
      const int c  = gw >> 5;          // s-chunk [0,32)
      const int hb = lane * 16;        // this lane's 16 contiguous h (64B aligned)
      float fcv[16], awv[16];
#pragma unroll
      for (int j = 0; j < 16; ++j) {
        fcv[j] = fch[b * HID + hb + j];
        awv[j] = att_w[hb + j];
      }
      float m = -1e30f, l = 0.f;
      float ctxa[16];
#pragma unroll
      for (int j = 0; j < 16; ++j) ctxa[j] = 0.f;

      for (int s = c; s < SFRM; s += NCHUNK) {
        const size_t row = ((size_t)s * BATCH + b) * HID + hb;
        const float4* ep4 = (const float4*)(encp + row);
        const float4* ev4 = (const float4*)(enc + row);
        float4 p0 = ep4[0], p1 = ep4[1], p2 = ep4[2], p3 = ep4[3];
        float4 e0 = ev4[0], e1 = ev4[1], e2 = ev4[2], e3 = ev4[3];
        const float epc[16] = {p0.x, p0.y, p0.z, p0.w, p1.x, p1.y, p1.z, p1.w,
                               p2.x, p2.y, p2.z, p2.w, p3.x, p3.y, p3.z, p3.w};
        const float env[16] = {e0.x, e0.y, e0.z, e0.w, e1.x, e1.y, e1.z, e1.w,
                               e2.x, e2.y, e2.z, e2.w, e3.x, e3.y, e3.z, e3.w};
        float part = 0.f;
#pragma unroll
        for (int j = 0; j < 16; ++j)
          part += TANHF(epc[j] + fcv[j]) * awv[j];
        const float e  = wred_sum(part);
        const float mn = fmaxf(m, e);
        const float sc = __expf(m - mn);
        const float w  = __expf(e - mn);
        l = l * sc + w;
#pragma unroll
        for (int j = 0; j < 16; ++j) ctxa[j] = ctxa[j] * sc + w * env[j];
        m = mn;
      }
      if (lane == 0) { pm[c * BATCH + b] = m; pl[c * BATCH + b] = l; }
      float* pc = pctx + ((size_t)c * BATCH + b) * HID + hb;
#pragma unroll
      for (int j = 0; j < 16; ++j) pc[j] = ctxa[j];
    }
    grid_sync(bar);

    // ---- Phase C: combine NCHUNK partials per b -> ctx[b][:] ----
    if (gw < BATCH) {
      const int b = gw;
      const float mc = pm[lane * BATCH + b];   // lane == chunk id
      const float lc = pl[lane * BATCH + b];
      const float mg  = wred_max(mc);
      const float wc  = __expf(mc - mg);
      const float lg  = wred_sum(lc * wc);
      const float inv = 1.f / lg;
      const int hb = lane * 16;
      float acc[16];
#pragma unroll
      for (int j = 0; j < 16; ++j) acc[j] = 0.f;
      for (int cc = 0; cc < NCHUNK; ++cc) {
        const float w = __shfl(wc, cc, 32) * inv;
        const float4* pc4 = (const float4*)(pctx + ((size_t)cc * BATCH + b) * HID + hb);
        float4 c0 = pc4[0], c1 = pc4[1], c2 = pc4[2], c3 = pc4[3];
        const float pcv[16] = {c0.x, c0.y, c0.z, c0.w, c1.x, c1.y, c1.z, c1.w,
                               c2.x, c2.y, c2.z, c2.w, c3.x, c3.y, c3.z, c3.w};
#pragma unroll
        for (int j = 0; j < 16; ++j) acc[j] += w * pcv[j];
      }
#pragma unroll
      for (int j = 0; j < 16; ++j) ctx[b * HID + hb + j] = acc[j];
    }
    grid_sync(bar);

    // ---- Phase D: GRU GEMMs: gi = [x|ctx] @ Wih^T, gh = h @ Whh^T ----
    {
      const float* xrow = (t >= 1) ? dec_in + (size_t)(t - 1) * BATCH * HID
                                   : nullptr;
      if (gw < 192) {                     // gi: 2 x 96 tiles of (32 x 1536)
        const int mt = gw & 1, nt = gw >> 1;
        gemm_tile_concat(xrow, ctx, Wih, mt * 16, nt * 16, gi, 3 * HID);
      } else if (gw < 384) {              // gh
        const int g2 = gw - 192;
        const int mt = g2 & 1, nt = g2 >> 1;
        gemm_tile_xwt<false>(hbuf, HID, Whh, HID, HID, mt * 16, nt * 16,
                             gh, 3 * HID, nullptr);
      }
    }
    grid_sync(bar);

    // ---- Phase E: GRU elementwise update: h <- (1-z)*n + z*h ----
    for (int idx = blockIdx.x * TPB + tid; idx < BATCH * HID; idx += NWG * TPB) {
      const int b = idx >> 9;             // /HID
      const int i = idx & (HID - 1);
      const size_t o = (size_t)b * 3 * HID + i;
      const float r = sigf(gi[o] + gh[o]);
      const float z = sigf(gi[o + HID] + gh[o + HID]);
      const float n = TANHF(gi[o + 2 * HID] + r * gh[o + 2 * HID]);
      hbuf[idx] = (1.f - z) * n + z * hbuf[idx];
    }
    grid_sync(bar);
  }
}

extern "C" void kernel_launch(void* const* d_in, const int* in_sizes, int n_in,
                              void* d_out, int out_size, void* d_ws,
                              size_t ws_size, hipStream_t stream) {
  (void)in_sizes; (void)n_in; (void)out_size; (void)ws_size;
  const float* dec_in = (const float*)d_in[0];
  const float* enc    = (const float*)d_in[1];
  const float* fc_W   = (const float*)d_in[2];
  const float* att_w  = (const float*)d_in[3];
  const float* Wih    = (const float*)d_in[4];
  const float* Whh    = (const float*)d_in[5];
  const float* spec_W = (const float*)d_in[6];
  const float* spec_b = (const float*)d_in[7];
  const float* gate_W = (const float*)d_in[8];
  const float* gate_b = (const float*)d_in[9];
  float* out = (float*)d_out;
  float* ws  = (float*)d_ws;

  unsigned* bar = (unsigned*)(ws + OFF_BAR);
  hipLaunchKernelGGL(init_bar_kernel, dim3(1), dim3(1), 0, stream, bar);
  hipLaunchKernelGGL(recurrent_decoder_kernel, dim3(NWG), dim3(TPB), 0, stream,
                     dec_in, enc, fc_W, att_w, Wih, Whh, spec_W, spec_b,
                     gate_W, gate_b, out, ws);
}